// Projection_73650099192097
// MI455X (gfx1250) — compile-verified
//
#include <hip/hip_runtime.h>

typedef __attribute__((ext_vector_type(2))) float v2f;
typedef __attribute__((ext_vector_type(4))) float v4f;
typedef __attribute__((ext_vector_type(8))) float v8f;

#define COLS 1024
#define OUT_DIM 512
#define ROWS_PER_WAVE 16
#define WAVES_PER_BLOCK 8
#define ROWS_PER_BLOCK (ROWS_PER_WAVE * WAVES_PER_BLOCK)

__global__ __launch_bounds__(256)
void proj_rowsum_wmma_kernel(const float* __restrict__ x,
                             float* __restrict__ out, int rows) {
    const int lane = threadIdx.x & 31;
    const int wave = threadIdx.x >> 5;
    const int rowBase = blockIdx.x * ROWS_PER_BLOCK + wave * ROWS_PER_WAVE;
    if (rowBase >= rows) return;   // uniform per wave -> EXEC stays all-ones

    // ---------------- Phase 1: row sums of x[tile, 512:1024] via WMMA ------
    // A (16x4 f32) layout: lanes 0-15 -> M=0..15 K={0,1}; lanes 16-31 -> K={2,3}.
    // B = all ones => D[m][n] = sum_k A[m][k] for every n (layout-robust in K).
    const int m    = lane & 15;
    const int koff = (lane >> 4) * 2;
    const float* rowp = x + (size_t)(rowBase + m) * COLS + OUT_DIM + koff;

    v2f ones = {1.0f, 1.0f};
    v8f acc0 = {}, acc1 = {}, acc2 = {}, acc3 = {};
#pragma unroll 2
    for (int k0 = 0; k0 < OUT_DIM; k0 += 16) {
        v2f a0 = *(const v2f*)(rowp + k0 + 0);
        v2f a1 = *(const v2f*)(rowp + k0 + 4);
        v2f a2 = *(const v2f*)(rowp + k0 + 8);
        v2f a3 = *(const v2f*)(rowp + k0 + 12);
        acc0 = __builtin_amdgcn_wmma_f32_16x16x4_f32(false, a0, false, ones,
                                                     (short)0, acc0, false, false);
        acc1 = __builtin_amdgcn_wmma_f32_16x16x4_f32(false, a1, false, ones,
                                                     (short)0, acc1, false, false);
        acc2 = __builtin_amdgcn_wmma_f32_16x16x4_f32(false, a2, false, ones,
                                                     (short)0, acc2, false, false);
        acc3 = __builtin_amdgcn_wmma_f32_16x16x4_f32(false, a3, false, ones,
                                                     (short)0, acc3, false, false);
    }
    v8f acc = (acc0 + acc1) + (acc2 + acc3);

    // D VGPR i: lanes 0-15 = rowsum(i), lanes 16-31 = rowsum(i+8). Broadcast.
    float s[16];
#pragma unroll
    for (int i = 0; i < 8; ++i) {
        s[i]     = __shfl(acc[i], 0, 32)  * (1.0f / OUT_DIM);
        s[i + 8] = __shfl(acc[i], 16, 32) * (1.0f / OUT_DIM);
    }

    // ---------------- Phase 2: out[r,:] = x[r,:512] + s[r] (streaming) -----
#pragma unroll
    for (int r = 0; r < ROWS_PER_WAVE; ++r) {
        const v4f* __restrict__ src = (const v4f*)(x + (size_t)(rowBase + r) * COLS);
        v4f* __restrict__ dst = (v4f*)(out + (size_t)(rowBase + r) * OUT_DIM);
        const float sv = s[r];
#pragma unroll
        for (int c = 0; c < OUT_DIM / 4; c += 32) {
            v4f v = __builtin_nontemporal_load(&src[c + lane]);
            v += sv;
            __builtin_nontemporal_store(v, &dst[c + lane]);
        }
    }
}

extern "C" void kernel_launch(void* const* d_in, const int* in_sizes, int n_in,
                              void* d_out, int out_size, void* d_ws, size_t ws_size,
                              hipStream_t stream) {
    const float* x = (const float*)d_in[0];
    float* out = (float*)d_out;
    const int rows = in_sizes[0] / COLS;                       // 131072
    const int blocks = (rows + ROWS_PER_BLOCK - 1) / ROWS_PER_BLOCK;  // 1024
    proj_rowsum_wmma_kernel<<<blocks, 256, 0, stream>>>(x, out, rows);
}